// Model_82875688944081
// MI455X (gfx1250) — compile-verified
//
#include <hip/hip_runtime.h>
#include <hip/hip_bf16.h>

#define B_SAMPLES 8192
#define T_TOKENS  1048576
#define NHID      256
#define NCLASSES  14
#define NPAD      16

typedef __attribute__((ext_vector_type(2))) float v2f;
typedef __attribute__((ext_vector_type(8))) float v8f;

// ---------------------------------------------------------------------------
// Kernel 1: exclusive prefix sum of lengths[B] -> offsets[B+1]
// Single block, 1024 threads, 8 elements/thread. Sums fit in int32 (T = 2^20).
// ---------------------------------------------------------------------------
__global__ void seg_scan_kernel(const int* __restrict__ lengths,
                                int* __restrict__ offsets) {
    __shared__ int ssum[1024];
    const int t = threadIdx.x;
    const int base = t * 8;
    int loc[8];
    int s = 0;
#pragma unroll
    for (int j = 0; j < 8; ++j) {
        loc[j] = lengths[base + j];
        s += loc[j];
    }
    ssum[t] = s;
    __syncthreads();
    // Hillis-Steele inclusive scan over the 1024 per-thread sums
    for (int off = 1; off < 1024; off <<= 1) {
        int v = (t >= off) ? ssum[t - off] : 0;
        __syncthreads();
        ssum[t] += v;
        __syncthreads();
    }
    int excl = ssum[t] - s;  // exclusive prefix for this thread's chunk
#pragma unroll
    for (int j = 0; j < 8; ++j) {
        offsets[base + j] = excl;
        excl += loc[j];
    }
    if (t == 1023) offsets[B_SAMPLES] = ssum[1023];
}

// ---------------------------------------------------------------------------
// Kernel 2: per-sample mean of gathered embedding rows, then tanh.
// One block per sample; 256 threads = one hidden dim each. Indices staged
// through LDS; each token row is a coalesced 1KB read (table lives in L2).
// ---------------------------------------------------------------------------
__global__ void seg_mean_tanh_kernel(const int* __restrict__ offsets,
                                     const int* __restrict__ indices,
                                     const float* __restrict__ emb,
                                     float* __restrict__ Amat) {
    __shared__ int sidx[256];
    const int tid = threadIdx.x;
    const int sample = blockIdx.x;
    const int s = offsets[sample];
    const int e = offsets[sample + 1];

    float acc = 0.0f;
    for (int base = s; base < e; base += 256) {
        const int n = (e - base < 256) ? (e - base) : 256;
        if (tid < n) sidx[tid] = indices[base + tid];
        __syncthreads();
        for (int j = 0; j < n; ++j) {
            // prefetch a few rows ahead (lowers to global_prefetch_b8)
            if (j + 8 < n) {
                __builtin_prefetch(emb + (size_t)sidx[j + 8] * NHID + tid, 0, 1);
            }
            acc += emb[(size_t)sidx[j] * NHID + tid];
        }
        __syncthreads();
    }

    const float len = (float)(e - s);
    const float m = (len > 0.0f) ? (acc / len) : 0.0f;
    Amat[(size_t)sample * NHID + tid] = tanhf(m);
}

// ---------------------------------------------------------------------------
// Kernel 3: out[B,14] = A[B,256] @ W[256,14] + b  via V_WMMA_F32_16X16X4_F32.
// W zero-padded to 16 columns in LDS. One wave per 16x16 output tile,
// 8 waves/block, 64 blocks -> 512 tiles covering all 8192 rows.
//
// A operand layout (32-bit A 16x4, ISA 7.12.2): lanes 0-15 hold M=lane,
// K = k+0 (v0) / k+1 (v1); lanes 16-31 hold M=lane-16, K = k+2 / k+3.
// B operand mirrors the K split: v0 = W[k+koff][n], v1 = W[k+koff+1][n].
// C/D: VGPR j -> M = j + 8*(lane/16), N = lane%16.
// ---------------------------------------------------------------------------
__global__ void wmma_head_kernel(const float* __restrict__ Amat,
                                 const float* __restrict__ W,
                                 const float* __restrict__ bias,
                                 float* __restrict__ out) {
    __shared__ float lw[NHID * NPAD];  // 16 KB padded weight

    const int tid = threadIdx.x;
    // cooperative load of W (256x14) into zero-padded 256x16 LDS tile
    for (int i = tid; i < NHID * NPAD; i += 256) {
        const int kk = i >> 4;
        const int nn = i & 15;
        lw[i] = (nn < NCLASSES) ? W[kk * NCLASSES + nn] : 0.0f;
    }
    __syncthreads();

    const int lane = tid & 31;
    const int wave = tid >> 5;
    const int tile = blockIdx.x * 8 + wave;   // 16-row tile of A
    const int row0 = tile * 16;

    const int mrow = lane & 15;
    const int koff = (lane >> 4) << 1;        // 0 for lanes 0-15, 2 for 16-31
    const float* __restrict__ arow = Amat + (size_t)(row0 + mrow) * NHID;

    v8f c = {};
#pragma unroll 4
    for (int k = 0; k < NHID; k += 4) {
        const float2 a2 = *(const float2*)(arow + k + koff);
        v2f av; av.x = a2.x; av.y = a2.y;
        v2f bv;
        bv.x = lw[(k + koff) * NPAD + mrow];
        bv.y = lw[(k + koff + 1) * NPAD + mrow];
        // 8 args: (neg_a, A, neg_b, B, c_mod, C, reuse_a, reuse_b)
        c = __builtin_amdgcn_wmma_f32_16x16x4_f32(
            false, av, false, bv, (short)0, c, false, false);
    }

    const int mhi = (lane >> 4) << 3;         // +8 for upper half-wave
    const int ncol = lane & 15;
    if (ncol < NCLASSES) {
        const float bb = bias[ncol];
#pragma unroll
        for (int j = 0; j < 8; ++j) {
            const int m = row0 + j + mhi;
            out[(size_t)m * NCLASSES + ncol] = c[j] + bb;
        }
    }
}

// ---------------------------------------------------------------------------
// Launch: inputs are (lengths[B], indices[T], emb[NTOKEN*256], W[256*14], b[14])
// Workspace: [0, 64KB)  -> offsets (B+1 ints)
//            [64KB, ..) -> A matrix tanh(means) [B,256] f32 (8 MB)
// ---------------------------------------------------------------------------
extern "C" void kernel_launch(void* const* d_in, const int* in_sizes, int n_in,
                              void* d_out, int out_size, void* d_ws, size_t ws_size,
                              hipStream_t stream) {
    const int*   lengths = (const int*)d_in[0];
    const int*   indices = (const int*)d_in[1];
    const float* emb     = (const float*)d_in[2];
    const float* W       = (const float*)d_in[3];
    const float* bias    = (const float*)d_in[4];
    float*       out     = (float*)d_out;

    int*   offsets = (int*)d_ws;
    float* Amat    = (float*)((char*)d_ws + (64 * 1024));

    seg_scan_kernel<<<1, 1024, 0, stream>>>(lengths, offsets);
    seg_mean_tanh_kernel<<<B_SAMPLES, 256, 0, stream>>>(offsets, indices, emb, Amat);
    wmma_head_kernel<<<B_SAMPLES / (16 * 8), 256, 0, stream>>>(Amat, W, bias, out);
}